// ExampleTiedDropout2_6786048327867
// MI455X (gfx1250) — compile-verified
//
#include <hip/hip_runtime.h>
#include <stdint.h>

typedef float v4f __attribute__((ext_vector_type(4)));

// ---------------------------------------------------------------------------
// Exact JAX threefry2x32 (Random123, 20 rounds).
// ---------------------------------------------------------------------------
__device__ __forceinline__ void threefry2x32(uint32_t k0, uint32_t k1,
                                             uint32_t x0, uint32_t x1,
                                             uint32_t& o0, uint32_t& o1) {
    const uint32_t ks2 = k0 ^ k1 ^ 0x1BD11BDAu;
    x0 += k0; x1 += k1;
#define TF_R4(r0, r1, r2, r3)                                        \
    x0 += x1; x1 = __builtin_rotateleft32(x1, r0); x1 ^= x0;         \
    x0 += x1; x1 = __builtin_rotateleft32(x1, r1); x1 ^= x0;         \
    x0 += x1; x1 = __builtin_rotateleft32(x1, r2); x1 ^= x0;         \
    x0 += x1; x1 = __builtin_rotateleft32(x1, r3); x1 ^= x0;
    TF_R4(13, 15, 26, 6)   x0 += k1;  x1 += ks2 + 1u;
    TF_R4(17, 29, 16, 24)  x0 += ks2; x1 += k0  + 2u;
    TF_R4(13, 15, 26, 6)   x0 += k0;  x1 += k1  + 3u;
    TF_R4(17, 29, 16, 24)  x0 += k1;  x1 += ks2 + 4u;
    TF_R4(13, 15, 26, 6)   x0 += ks2; x1 += k0  + 5u;
#undef TF_R4
    o0 = x0; o1 = x1;
}

// Problem constants (from the reference).
#define SEED_U32     101010u
#define B_DIM        256
#define C_DIM        256
#define HW_DIM       1024            // 32*32
#define EX_STRIDE    262144u         // C*H*W
#define C_FIXED      51              // int(0.2*256)
#define MEM_BASE     52224u          // C_FIXED*HW
#define N_MEM        209920u         // (C-C_FIXED)*HW
#define HALF_MEM     104960u         // N_MEM/2
#define PAIR_V4      26240u          // HALF_MEM/4

// ---------------------------------------------------------------------------
// Kernel 0: per-example key fold-in.  keys[b] = threefry((0,SEED), (0,idx[b])).
// ---------------------------------------------------------------------------
__global__ void tied_dropout_fold_keys(const int* __restrict__ idx,
                                       uint32_t* __restrict__ keys, int nB) {
    int i = blockIdx.x * blockDim.x + threadIdx.x;
    if (i >= nB) return;
    uint32_t o0, o1;
    threefry2x32(0u, SEED_U32, 0u, (uint32_t)idx[i], o0, o1);
    keys[2 * i + 0] = o0;
    keys[2 * i + 1] = o1;
}

// ---------------------------------------------------------------------------
// Kernel 1: fixed channels (pure memcpy, 20% of traffic) via CDNA5 async
// global->LDS->global pipe (ASYNCcnt).  One block = one 4KB channel plane;
// each lane bounces 16B through its private LDS slot: no VGPR data movement.
// grid = (C_FIXED, B), block = 256 (8 wave32s).
// ---------------------------------------------------------------------------
__global__ void tied_dropout_copy_fixed(const float* __restrict__ X,
                                        float* __restrict__ Y) {
    __shared__ float smem[HW_DIM];       // 4 KB
    const unsigned t = threadIdx.x;      // 0..255, one float4 each
    const size_t off = (size_t)blockIdx.y * EX_STRIDE +
                       (size_t)blockIdx.x * HW_DIM + (size_t)t * 4u;

    // Low 32 bits of the generic address of an LDS object are the LDS byte
    // offset (ISA 10.2 aperture mapping).
    uint32_t lds = (uint32_t)(uintptr_t)(&smem[t * 4u]);
    unsigned long long src = (unsigned long long)(uintptr_t)(X + off);
    unsigned long long dst = (unsigned long long)(uintptr_t)(Y + off);

    asm volatile("global_load_async_to_lds_b128 %0, %1, off"
                 :: "v"(lds), "v"(src) : "memory");
    asm volatile("s_wait_asynccnt 0" ::: "memory");
    asm volatile("global_store_async_from_lds_b128 %0, %1, off"
                 :: "v"(dst), "v"(lds) : "memory");
    // s_endpgm performs an implicit wait-idle; nothing reuses this LDS.
}

// ---------------------------------------------------------------------------
// Kernel 2: memorization channels.  JAX counter mode pairs element e with
// e+HALF_MEM in a single threefry call (out0/out1), so each thread processes
// two float4s (8 elements) with 4 threefry calls.  Streaming data uses
// non-temporal b128 loads/stores (tensor > L2, touched once).
// grid = (ceil(PAIR_V4/256), B), block = 256.
// ---------------------------------------------------------------------------
__global__ void tied_dropout_mask_mem(const float* __restrict__ X,
                                      float* __restrict__ Y,
                                      const uint32_t* __restrict__ keys) {
    const uint32_t j = blockIdx.x * blockDim.x + threadIdx.x;   // float4-pair id
    if (j >= PAIR_V4) return;
    const uint32_t b  = blockIdx.y;
    const uint32_t k0 = keys[2 * b + 0];
    const uint32_t k1 = keys[2 * b + 1];

    const uint32_t e0   = 4u * j;
    const size_t   base = (size_t)b * EX_STRIDE + MEM_BASE;

    v4f x0 = __builtin_nontemporal_load((const v4f*)(X + base + e0));
    v4f x1 = __builtin_nontemporal_load((const v4f*)(X + base + e0 + HALF_MEM));
    v4f y0, y1;

#pragma unroll
    for (int i = 0; i < 4; ++i) {
        uint32_t o0, o1;
        threefry2x32(k0, k1, e0 + (uint32_t)i, e0 + (uint32_t)i + HALF_MEM,
                     o0, o1);
        // JAX uniform: bitcast((bits>>9)|0x3f800000) - 1.0 in [0,1); keep if < p.
        float u0 = __builtin_bit_cast(float, (o0 >> 9) | 0x3f800000u) - 1.0f;
        float u1 = __builtin_bit_cast(float, (o1 >> 9) | 0x3f800000u) - 1.0f;
        y0[i] = (u0 < 0.1f) ? x0[i] : 0.0f;
        y1[i] = (u1 < 0.1f) ? x1[i] : 0.0f;
    }

    __builtin_nontemporal_store(y0, (v4f*)(Y + base + e0));
    __builtin_nontemporal_store(y1, (v4f*)(Y + base + e0 + HALF_MEM));
}

// ---------------------------------------------------------------------------
// Host launcher.
// Inputs: d_in[0] = X (float32, [256,256,32,32]), d_in[1] = idx (int, [256]),
//         d_in[2] = epoch (ignored; reference implements the epoch==0 branch).
// d_ws: first 2*B uint32 hold the folded per-example keys.
// ---------------------------------------------------------------------------
extern "C" void kernel_launch(void* const* d_in, const int* in_sizes, int n_in,
                              void* d_out, int out_size, void* d_ws, size_t ws_size,
                              hipStream_t stream) {
    const float* X   = (const float*)d_in[0];
    const int*   idx = (const int*)d_in[1];
    float*       Y   = (float*)d_out;
    uint32_t*    keys = (uint32_t*)d_ws;

    const int nB = in_sizes[1];          // 256 examples

    // 0) fold per-example keys (tiny)
    tied_dropout_fold_keys<<<(nB + 255) / 256, 256, 0, stream>>>(idx, keys, nB);

    // 1) fixed channels: async LDS-bounce memcpy
    tied_dropout_copy_fixed<<<dim3(C_FIXED, nB), 256, 0, stream>>>(X, Y);

    // 2) memorization channels: threefry mask + NT streaming
    const unsigned nblk = (PAIR_V4 + 255u) / 256u;   // 103
    tied_dropout_mask_mem<<<dim3(nblk, nB), 256, 0, stream>>>(X, Y, keys);
}